// SparseModel_89618787598437
// MI455X (gfx1250) — compile-verified
//
#include <hip/hip_runtime.h>
#include <hip/hip_bf16.h>
#include <math.h>

typedef float v2f __attribute__((ext_vector_type(2)));
typedef float v8f __attribute__((ext_vector_type(8)));

#define B_TOT   16384
#define IN_DIM  512
#define NH      128
#define NO      256
#define SRC_D   (IN_DIM + NH)        // 640
#define TB      32                   // batch rows per block
#define XP      514                  // padded x pitch (dwords): 514%64==2 -> conflict-free, 8B aligned rows
#define HP      33                   // padded H pitch (odd -> conflict-free everywhere)
#define OPP     33                   // padded OUT pitch

// LDS layout (float offsets)
#define X_OFF   0
#define H_OFF   (TB * XP)                      // 16448
#define O_OFF   (H_OFF + NH * HP)              // 20672
#define LDS_FLOATS (O_OFF + NO * OPP)          // 29120
#define LDS_BYTES  (LDS_FLOATS * 4)            // 116480 B

// workspace layout: W1 [NH][IN_DIM] f32, W2 [NO][SRC_D] f32, rp_h[NH+1], rp_o[NO+1]
#define W1_FLOATS (NH * IN_DIM)                // 65536
#define W2_FLOATS (NO * SRC_D)                 // 163840
#define W_FLOATS  (W1_FLOATS + W2_FLOATS)

// ---------------- prep kernels (run every launch; deterministic) ----------------

__global__ void k_zero(float* __restrict__ w) {
    int i = blockIdx.x * 256 + threadIdx.x;
    if (i < W_FLOATS) w[i] = 0.f;
}

// scatter identity-edge weights into the dense matrix (unique positions -> plain stores)
__global__ void k_densify(const int* __restrict__ rows, const int* __restrict__ cols,
                          const int* __restrict__ acts, const float* __restrict__ w,
                          int E, int pitch, float* __restrict__ W) {
    int e = blockIdx.x * 256 + threadIdx.x;
    if (e < E && acts[e] == 1) W[(size_t)rows[e] * pitch + cols[e]] = w[e];
}

// rows[] is nondecreasing (np.nonzero row-major) -> CSR row pointers
__global__ void k_rowptr(const int* __restrict__ rows, int E, int NR, int* __restrict__ rp) {
    int e = blockIdx.x * 256 + threadIdx.x;
    if (e >= E) return;
    int r = rows[e];
    int prev = (e == 0) ? -1 : rows[e - 1];
    for (int rr = prev + 1; rr <= r; ++rr) rp[rr] = e;
    if (e == E - 1)
        for (int rr = r + 1; rr <= NR; ++rr) rp[rr] = E;
}

// ---------------- helpers ----------------

__device__ __forceinline__ float edge_act(float z, int a) {
    float r = z;
    if (a == 2)      r = tanhf(z);
    else if (a == 3) r = fmaxf(z, 0.f);
    else if (a == 4) r = 1.f / (1.f + __expf(-z));
    return r;
}

__device__ __forceinline__ int   sfl_i(int v)   { return __builtin_amdgcn_readfirstlane(v); }
__device__ __forceinline__ float sfl_f(float v) {
    return __uint_as_float((unsigned)__builtin_amdgcn_readfirstlane((int)__float_as_uint(v)));
}

// ---------------- main fused kernel ----------------

__global__ __launch_bounds__(256) void k_wann(
    const float* __restrict__ x,
    const float* __restrict__ W1, const float* __restrict__ W2,
    const int* __restrict__ rows_h, const int* __restrict__ cols_h,
    const int* __restrict__ acts_h, const float* __restrict__ wh,
    const int* __restrict__ rows_o, const int* __restrict__ cols_o,
    const int* __restrict__ acts_o, const float* __restrict__ wo,
    const int* __restrict__ rp_h, const int* __restrict__ rp_o,
    float* __restrict__ out)
{
    extern __shared__ float lds[];
    const int t    = threadIdx.x;
    const int wv   = sfl_i(t >> 5);        // wave id, force-scalar (compiler can't prove it)
    const int lane = t & 31;
    const int b0   = blockIdx.x * TB;

    // ---- stage x tile [32][512] into LDS via async Global->LDS (ASYNCcnt path) ----
    // b64 granules keep the 514-dword padded pitch (8B-aligned rows, bank-conflict-free).
    {
        const float* xg = x + (size_t)b0 * IN_DIM;
        for (int i = t; i < TB * (IN_DIM / 2); i += 256) {   // 8192 x 8B chunks
            int b  = i >> 8;                  // IN_DIM/2 == 256
            int c2 = i & 255;
            unsigned ldsoff = (unsigned)(size_t)(lds + b * XP + c2 * 2); // flat addr[31:0] == LDS offset
            unsigned long long ga = (unsigned long long)(size_t)(xg + (size_t)b * IN_DIM + c2 * 2);
            asm volatile("global_load_async_to_lds_b64 %0, %1, off"
                         :: "v"(ldsoff), "v"(ga) : "memory");
        }
        asm volatile("s_wait_asynccnt 0" ::: "memory");
    }
    __syncthreads();

    const int l15   = lane & 15;
    const int khalf = (lane >> 4) << 1;     // 0 or 2 (K sub-offset per ISA A/B layout)

    // ---- hidden layer, identity-edge part: H_lin = Xtile * W1^T via f32 WMMA ----
    // wave w owns N-tile n0 = 16*w (8 waves cover NH=128), M-tiles 0 and 1.
    {
        v8f acc0 = {0.f,0.f,0.f,0.f,0.f,0.f,0.f,0.f};
        v8f acc1 = acc0;
        const int n0 = wv * 16;
        const float* wrow = W1 + (size_t)(n0 + l15) * IN_DIM + khalf;
        const float* arow = lds + l15 * XP + khalf;
        #pragma unroll 4
        for (int k = 0; k < IN_DIM; k += 4) {
            v2f bf, a0, a1;
            { float2 v = *(const float2*)(wrow + k);            bf.x = v.x; bf.y = v.y; }
            { float2 v = *(const float2*)(arow + k);            a0.x = v.x; a0.y = v.y; }
            { float2 v = *(const float2*)(arow + 16 * XP + k);  a1.x = v.x; a1.y = v.y; }
            acc0 = __builtin_amdgcn_wmma_f32_16x16x4_f32(false, a0, false, bf, (short)0, acc0, false, false);
            acc1 = __builtin_amdgcn_wmma_f32_16x16x4_f32(false, a1, false, bf, (short)0, acc1, false, false);
        }
        float* HT = lds + H_OFF;
        const int nn = n0 + l15;
        const int mb = (lane >> 4) << 3;
        #pragma unroll
        for (int i = 0; i < 8; ++i) {
            HT[nn * HP + mb + i]      = acc0[i];   // batch rows 0..15
            HT[nn * HP + 16 + mb + i] = acc1[i];   // batch rows 16..31
        }
    }
    __syncthreads();

    // ---- hidden layer, nonlinear edges: wave w owns H rows [16w,16w+16), lane = batch ----
    // Edge stream is wave-uniform: scalarize metadata so loop control/branches live in SALU.
    {
        float* HT = lds + H_OFF;
        const int e0 = sfl_i(rp_h[wv * 16]);
        const int e1 = sfl_i(rp_h[wv * 16 + 16]);
        int cur = -1; float acc = 0.f;
        for (int e = e0; e < e1; ++e) {
            int r = sfl_i(rows_h[e]);
            if (r != cur) {
                if (cur >= 0) HT[cur * HP + lane] += acc;
                cur = r; acc = 0.f;
            }
            int a = sfl_i(acts_h[e]);
            if (a != 1) {                          // identity edges already in WMMA part
                int   c = sfl_i(cols_h[e]);
                float w = sfl_f(wh[e]);
                acc += edge_act(lds[lane * XP + c] * w, a);
            }
        }
        if (cur >= 0) HT[cur * HP + lane] += acc;
    }
    __syncthreads();

    // ---- output layer, identity-edge part: OUT_lin = [X|H] * W2^T via WMMA ----
    // wave w owns N-tiles {w, w+8} (covers NO=256), M-tiles 0 and 1; K spans 640.
    {
        const float* HT = lds + H_OFF;
        v8f o00 = {0.f,0.f,0.f,0.f,0.f,0.f,0.f,0.f};
        v8f o01 = o00, o10 = o00, o11 = o00;
        const int nA = wv * 16 + l15;
        const int nB = (wv + 8) * 16 + l15;
        const float* w2a = W2 + (size_t)nA * SRC_D + khalf;
        const float* w2b = W2 + (size_t)nB * SRC_D + khalf;
        const float* arow = lds + l15 * XP + khalf;
        for (int k = 0; k < SRC_D; k += 4) {
            v2f a0, a1;
            if (k < IN_DIM) {
                float2 v0 = *(const float2*)(arow + k);
                float2 v1 = *(const float2*)(arow + 16 * XP + k);
                a0.x = v0.x; a0.y = v0.y; a1.x = v1.x; a1.y = v1.y;
            } else {
                int h = k - IN_DIM + khalf;
                a0.x = HT[h * HP + l15];           a0.y = HT[(h + 1) * HP + l15];
                a1.x = HT[h * HP + 16 + l15];      a1.y = HT[(h + 1) * HP + 16 + l15];
            }
            v2f bfa, bfb;
            { float2 v = *(const float2*)(w2a + k); bfa.x = v.x; bfa.y = v.y; }
            { float2 v = *(const float2*)(w2b + k); bfb.x = v.x; bfb.y = v.y; }
            o00 = __builtin_amdgcn_wmma_f32_16x16x4_f32(false, a0, false, bfa, (short)0, o00, false, false);
            o01 = __builtin_amdgcn_wmma_f32_16x16x4_f32(false, a1, false, bfa, (short)0, o01, false, false);
            o10 = __builtin_amdgcn_wmma_f32_16x16x4_f32(false, a0, false, bfb, (short)0, o10, false, false);
            o11 = __builtin_amdgcn_wmma_f32_16x16x4_f32(false, a1, false, bfb, (short)0, o11, false, false);
        }
        float* OT = lds + O_OFF;
        const int mb = (lane >> 4) << 3;
        #pragma unroll
        for (int i = 0; i < 8; ++i) {
            OT[nA * OPP + mb + i]      = o00[i];
            OT[nA * OPP + 16 + mb + i] = o01[i];
            OT[nB * OPP + mb + i]      = o10[i];
            OT[nB * OPP + 16 + mb + i] = o11[i];
        }
    }
    __syncthreads();

    // ---- output layer, nonlinear edges: wave w owns OUT rows [32w,32w+32) ----
    {
        const float* HT = lds + H_OFF;
        float* OT = lds + O_OFF;
        const int e0 = sfl_i(rp_o[wv * 32]);
        const int e1 = sfl_i(rp_o[wv * 32 + 32]);
        int cur = -1; float acc = 0.f;
        for (int e = e0; e < e1; ++e) {
            int r = sfl_i(rows_o[e]);
            if (r != cur) {
                if (cur >= 0) OT[cur * OPP + lane] += acc;
                cur = r; acc = 0.f;
            }
            int a = sfl_i(acts_o[e]);
            if (a != 1) {
                int   c = sfl_i(cols_o[e]);
                float w = sfl_f(wo[e]);
                float s = (c < IN_DIM) ? lds[lane * XP + c] : HT[(c - IN_DIM) * HP + lane];
                acc += edge_act(s * w, a);
            }
        }
        if (cur >= 0) OT[cur * OPP + lane] += acc;
    }
    __syncthreads();

    // ---- final tanh + coalesced store (thread t = output column) ----
    {
        const float* OT = lds + O_OFF;
        float* op = out + (size_t)b0 * NO + t;
        #pragma unroll 4
        for (int b = 0; b < TB; ++b)
            op[(size_t)b * NO] = tanhf(OT[t * OPP + b]);
    }
}

// ---------------- launcher ----------------

extern "C" void kernel_launch(void* const* d_in, const int* in_sizes, int n_in,
                              void* d_out, int out_size, void* d_ws, size_t ws_size,
                              hipStream_t stream) {
    const float* x      = (const float*)d_in[0];
    const float* wh     = (const float*)d_in[1];
    const float* wo     = (const float*)d_in[2];
    const int*   rows_h = (const int*)d_in[3];
    const int*   cols_h = (const int*)d_in[4];
    const int*   acts_h = (const int*)d_in[5];
    const int*   rows_o = (const int*)d_in[6];
    const int*   cols_o = (const int*)d_in[7];
    const int*   acts_o = (const int*)d_in[8];
    float* out = (float*)d_out;
    const int Eh = in_sizes[1];
    const int Eo = in_sizes[2];

    float* W1  = (float*)d_ws;
    float* W2  = W1 + W1_FLOATS;
    int*   rp_h = (int*)(W2 + W2_FLOATS);
    int*   rp_o = rp_h + (NH + 1);

    k_zero<<<(W_FLOATS + 255) / 256, 256, 0, stream>>>(W1);
    k_densify<<<(Eh + 255) / 256, 256, 0, stream>>>(rows_h, cols_h, acts_h, wh, Eh, IN_DIM, W1);
    k_densify<<<(Eo + 255) / 256, 256, 0, stream>>>(rows_o, cols_o, acts_o, wo, Eo, SRC_D, W2);
    k_rowptr<<<(Eh + 255) / 256, 256, 0, stream>>>(rows_h, Eh, NH, rp_h);
    k_rowptr<<<(Eo + 255) / 256, 256, 0, stream>>>(rows_o, Eo, NO, rp_o);

    (void)hipFuncSetAttribute((const void*)k_wann,
                              hipFuncAttributeMaxDynamicSharedMemorySize, LDS_BYTES);
    k_wann<<<B_TOT / TB, 256, LDS_BYTES, stream>>>(
        x, W1, W2,
        rows_h, cols_h, acts_h, wh,
        rows_o, cols_o, acts_o, wo,
        rp_h, rp_o, out);
}